// MFF_38809324487316
// MI455X (gfx1250) — compile-verified
//
#include <hip/hip_runtime.h>
#include <hip/hip_bf16.h>

typedef float v2f __attribute__((ext_vector_type(2)));
typedef float v8f __attribute__((ext_vector_type(8)));

#define BATCH 16
#define CH    256
#define HALF  128
#define HW    6400
#define BN_EPS 1e-5f

// ---------------------------------------------------------------------------
// Kernel 1: per-(b,c) spatial mean of x1 -> means[16*256]
// ---------------------------------------------------------------------------
__global__ void MFF_means_kernel(const float* __restrict__ x1,
                                 float* __restrict__ means) {
    const int bc = blockIdx.x;                 // 0..4095
    const float* p = x1 + (size_t)bc * HW;
    float s = 0.f;
    for (int i = threadIdx.x; i < HW; i += 256) s += p[i];
    __shared__ float red[256];
    red[threadIdx.x] = s;
    __syncthreads();
    for (int off = 128; off > 0; off >>= 1) {
        if (threadIdx.x < off) red[threadIdx.x] += red[threadIdx.x + off];
        __syncthreads();
    }
    if (threadIdx.x == 0) means[bc] = red[0] * (1.0f / (float)HW);
}

// ---------------------------------------------------------------------------
// Kernel 2: ECA conv1d(k=5, pad=2) + sigmoid + top-128 selection.
// pos_idx[b][0..127] = selected channels ascending (matches the reference's
// top_k -> mask -> stable argsort).
// ---------------------------------------------------------------------------
__global__ void MFF_select_kernel(const float* __restrict__ means,
                                  const float* __restrict__ eca_w,
                                  int* __restrict__ pos_idx) {
    const int b = blockIdx.x;
    const int c = threadIdx.x;                 // 256 threads = 256 channels
    __shared__ float sm[CH];
    __shared__ float sc[CH];
    __shared__ int   smem[CH];
    sm[c] = means[b * CH + c];
    __syncthreads();
    float y = 0.f;
    #pragma unroll
    for (int j = 0; j < 5; ++j) {
        const int cc = c - 2 + j;
        const float v = (cc >= 0 && cc < CH) ? sm[cc] : 0.f;
        y += eca_w[j] * v;
    }
    const float s = 1.0f / (1.0f + __expf(-y));
    sc[c] = s;
    __syncthreads();
    int rank = 0;
    for (int j = 0; j < CH; ++j) {
        const float sj = sc[j];
        rank += (sj > s) || (sj == s && j < c);   // lax.top_k tie rule
    }
    const int member = (rank < HALF) ? 1 : 0;
    smem[c] = member;
    __syncthreads();
    if (member) {
        int p = 0;
        for (int j = 0; j < c; ++j) p += smem[j];
        pos_idx[b * HALF + p] = c;
    }
}

// ---------------------------------------------------------------------------
// Kernel 3: fused gather + 127x129 1x1-conv GEMM (f32 WMMA) + BN + leaky +
// residual add + x0 passthrough. One wave per 16-pixel tile, 8 waves/block.
// conv_w staged in 64KB LDS, XOR-swizzled (pitch 128, col ^= 4*(row&15)) so
// the 16x4 A-fragment reads are conflict-free b64 loads for 4-aligned K.
// Epilogue is fused per M-tile so accumulators die early (VGPR pressure).
// ---------------------------------------------------------------------------
__global__ __launch_bounds__(256) void MFF_main_kernel(
        const float* __restrict__ x0, const float* __restrict__ x1,
        const float* __restrict__ conv_w,
        const float* __restrict__ bn_gamma, const float* __restrict__ bn_beta,
        const float* __restrict__ bn_mean,  const float* __restrict__ bn_var,
        const int* __restrict__ pos_idx, float* __restrict__ out) {
    __shared__ float sW[128 * 128];            // exactly 64 KB

    const int tid = threadIdx.x;
    // Stage conv_w cols 0..127 (col 128 = neg-mean coeff handled in the
    // epilogue), pad row 127 with zeros.
    for (int idx = tid; idx < 128 * 128; idx += 256) {
        const int m = idx >> 7, k = idx & 127;
        const float v = (m < 127) ? conv_w[m * 129 + k] : 0.f;
        sW[(m << 7) | (k ^ ((m & 15) << 2))] = v;
    }
    __syncthreads();

    const int b      = blockIdx.x / 50;
    const int wave   = tid >> 5;
    const int lane   = tid & 31;
    const int l      = lane & 15;              // pixel / M-row within tile
    const int hi     = lane >> 4;              // half-wave selector
    const int pix0   = ((blockIdx.x % 50) * 8 + wave) * 16;

    const float* x1b  = x1 + (size_t)b * CH * HW;
    const int*   posb = pos_idx + b * HALF;

    // ---- Phase 0: per-pixel total sum over all 256 channels (float4) ----
    float4 acc4 = make_float4(0.f, 0.f, 0.f, 0.f);
    {
        const float* base = x1b + pix0 + (lane & 3) * 4;
        #pragma unroll 4
        for (int c = (lane >> 2); c < CH; c += 8) {
            const float4 v = *(const float4*)(base + (size_t)c * HW);
            acc4.x += v.x; acc4.y += v.y; acc4.z += v.z; acc4.w += v.w;
        }
        for (int off = 4; off < 32; off <<= 1) {
            acc4.x += __shfl_xor(acc4.x, off, 32);
            acc4.y += __shfl_xor(acc4.y, off, 32);
            acc4.z += __shfl_xor(acc4.z, off, 32);
            acc4.w += __shfl_xor(acc4.w, off, 32);
        }
    }
    // Per-lane total for pixel p = l: source lane p>>2, component p&3.
    float totsum;
    {
        const float t0 = __shfl(acc4.x, l >> 2, 32);
        const float t1 = __shfl(acc4.y, l >> 2, 32);
        const float t2 = __shfl(acc4.z, l >> 2, 32);
        const float t3 = __shfl(acc4.w, l >> 2, 32);
        const int pc = l & 3;
        totsum = (pc == 0) ? t0 : (pc == 1) ? t1 : (pc == 2) ? t2 : t3;
    }

    // ---- Phase 1: K=128 GEMM via v_wmma_f32_16x16x4_f32, 8 M-tiles ----
    v8f acc[8];
    #pragma unroll
    for (int t = 0; t < 8; ++t) acc[t] = (v8f){0.f,0.f,0.f,0.f,0.f,0.f,0.f,0.f};

    float psum = 0.f;
    for (int i = 0; i < HALF; i += 4) {
        const int kA = i + 2 * hi;             // this lane's two K indices (even)
        const int cA = posb[kA];
        const int cB = posb[kA + 1];
        v2f bfrag;
        const float v0 = x1b[(size_t)cA * HW + pix0 + l];
        const float v1 = x1b[(size_t)cB * HW + pix0 + l];
        bfrag[0] = v0; bfrag[1] = v1;
        psum += v0 + v1;

        // kA even => swizzled cols (kA^(4l), (kA+1)^(4l)) are adjacent:
        // load the whole 8-byte A fragment with one aligned ds_load_b64.
        const int c0 = kA ^ (l << 2);
        #pragma unroll
        for (int t = 0; t < 8; ++t) {
            const int rowbase = (t * 16 + l) << 7;
            const v2f afrag = *(const v2f*)(sW + (rowbase | c0));
            acc[t] = __builtin_amdgcn_wmma_f32_16x16x4_f32(
                false, afrag, false, bfrag, (short)0, acc[t], false, false);
        }
    }
    psum += __shfl_xor(psum, 16, 32);
    const float neg_mean = (totsum - psum) * (1.0f / (float)HALF);

    float*       outb = out + (size_t)b * (2 * CH) * HW;
    const float* x0b  = x0 + (size_t)b * CH * HW;

    // ---- Fused epilogue: neg-mean column + BN + LeakyReLU + residual,
    //      stored tile-by-tile so acc[t] dies right after tile t ----
    #pragma unroll
    for (int t = 0; t < 8; ++t) {
        #pragma unroll
        for (int r = 0; r < 8; ++r) {
            const int m = t * 16 + r + hi * 8;  // C-layout: lanes16-31 -> M+8
            if (m < 127) {
                float z = acc[t][r] + conv_w[m * 129 + 128] * neg_mean;
                const float sc = bn_gamma[m] * __frsqrt_rn(bn_var[m] + BN_EPS);
                z = (z - bn_mean[m]) * sc + bn_beta[m];
                z = (z > 0.f) ? z : 0.1f * z;
                outb[(size_t)(CH + 129 + m) * HW + pix0 + l] =
                    x1b[(size_t)(129 + m) * HW + pix0 + l] + z;
            }
        }
    }

    // ---- out[:, 256+o] = x1[o] + tmp1[o], o = 0..127 (gathered pos) ----
    #pragma unroll 4
    for (int j = 0; j < 64; ++j) {
        const int o = 2 * j + hi;
        const float a = x1b[(size_t)o * HW + pix0 + l];
        const float g = x1b[(size_t)posb[o] * HW + pix0 + l];
        outb[(size_t)(CH + o) * HW + pix0 + l] = a + g;
    }
    // o = 128: neg-mean channel
    if (hi == 0) {
        outb[(size_t)(CH + 128) * HW + pix0 + l] =
            x1b[(size_t)128 * HW + pix0 + l] + neg_mean;
    }

    // ---- out[:, 0:256] = x0 (float4 streaming copy) ----
    #pragma unroll 4
    for (int cc = 0; cc < 32; ++cc) {
        const int c = cc * 8 + (lane >> 2);
        const size_t off = (size_t)c * HW + pix0 + (lane & 3) * 4;
        *(float4*)(outb + off) = *(const float4*)(x0b + off);
    }
}

// ---------------------------------------------------------------------------
extern "C" void kernel_launch(void* const* d_in, const int* in_sizes, int n_in,
                              void* d_out, int out_size, void* d_ws, size_t ws_size,
                              hipStream_t stream) {
    const float* x0       = (const float*)d_in[0];
    const float* x1       = (const float*)d_in[1];
    const float* eca_w    = (const float*)d_in[2];
    const float* conv_w   = (const float*)d_in[3];
    const float* bn_gamma = (const float*)d_in[4];
    const float* bn_beta  = (const float*)d_in[5];
    const float* bn_mean  = (const float*)d_in[6];
    const float* bn_var   = (const float*)d_in[7];
    float* out = (float*)d_out;

    float* means = (float*)d_ws;                                     // 16 KB
    int*   pos   = (int*)((char*)d_ws + BATCH * CH * sizeof(float)); // 8 KB

    MFF_means_kernel<<<BATCH * CH, 256, 0, stream>>>(x1, means);
    MFF_select_kernel<<<BATCH, 256, 0, stream>>>(means, eca_w, pos);
    MFF_main_kernel<<<BATCH * 50, 256, 0, stream>>>(
        x0, x1, conv_w, bn_gamma, bn_beta, bn_mean, bn_var, pos, out);
}